// GCNEncoder_10264971837864
// MI455X (gfx1250) — compile-verified
//
#include <hip/hip_runtime.h>

// ---------------------------------------------------------------------------
// GCN encoder, fp32 end-to-end on gfx1250 (wave32).
//   deg -> dinv -> [WMMA f32 GEMM -> zero -> edge scatter (f32 atomics)
//                   -> selfloop+bias(+relu)] x 2 layers
// ---------------------------------------------------------------------------

typedef float v2f __attribute__((ext_vector_type(2)));
typedef float v8f __attribute__((ext_vector_type(8)));

#define N_NODES 100000
#define IN_DIM  64
#define HID_DIM 128
#define EMB_DIM 64

// ---------------------------------- utility kernels ------------------------

__global__ void zero_f32(float* __restrict__ p, int n) {
    int i = blockIdx.x * blockDim.x + threadIdx.x;
    if (i < n) p[i] = 0.0f;
}

// self loop contributes 1 to every node's degree
__global__ void init_deg(float* __restrict__ deg, int n) {
    int i = blockIdx.x * blockDim.x + threadIdx.x;
    if (i < n) deg[i] = 1.0f;
}

__global__ void edge_deg(const long long* __restrict__ col,
                         float* __restrict__ deg, int nEdges) {
    int e = blockIdx.x * blockDim.x + threadIdx.x;
    if (e < nEdges) unsafeAtomicAdd(&deg[(int)col[e]], 1.0f);
}

__global__ void deg_to_dinv(float* __restrict__ deg, int n) {
    int i = blockIdx.x * blockDim.x + threadIdx.x;
    if (i < n) deg[i] = rsqrtf(deg[i]);   // deg >= 1 always (self loop)
}

// ---------------------------------- WMMA GEMM ------------------------------
// D[M,N] = A[M,K] @ B[K,N], fp32 via V_WMMA_F32_16X16X4_F32.
// One wave computes one 16x16 tile; blockDim.x = 32 * (N/16); grid = M/16.
template <int K, int N>
__global__ void gemm_wmma_f32(const float* __restrict__ A,
                              const float* __restrict__ B,
                              float* __restrict__ D) {
    const int wave  = threadIdx.x >> 5;
    const int lane  = threadIdx.x & 31;
    const int lmod  = lane & 15;     // M (for A) / N (for B,D) position
    const int lhalf = lane >> 4;     // which K pair / which M half

    const int row0 = blockIdx.x * 16;
    const int col0 = wave * 16;

    v8f acc = {};
    const float* arow = A + (size_t)(row0 + lmod) * K;

#pragma unroll
    for (int k = 0; k < K; k += 4) {
        // A 16x4 f32: lane(0-15) holds K=k+0,k+1 ; lane(16-31) holds K=k+2,k+3
        v2f a = *(const v2f*)(arow + k + lhalf * 2);
        // B 4x16 f32 (mirror layout): VGPR0 = row kb, VGPR1 = row kb+1
        const int kb = k + lhalf * 2;
        v2f b;
        b.x = B[(size_t)kb * N + col0 + lmod];
        b.y = B[(size_t)(kb + 1) * N + col0 + lmod];
        acc = __builtin_amdgcn_wmma_f32_16x16x4_f32(
            /*neg_a=*/false, a, /*neg_b=*/false, b,
            /*c_mod=*/(short)0, acc, /*reuse_a=*/false, /*reuse_b=*/false);
    }

    // C/D layout: VGPR v holds M = v + lhalf*8, N = lmod
    const int rbase = row0 + lhalf * 8;
#pragma unroll
    for (int v = 0; v < 8; ++v)
        D[(size_t)(rbase + v) * N + col0 + lmod] = acc[v];
}

// ---------------------------------- edge scatter ---------------------------
// One wave per edge: gather src[row]*norm, atomic-add into dst[col].
// DIM=128 -> float4 per lane; DIM=64 -> float2 per lane.
template <int DIM>
__global__ void scatter_edges(const long long* __restrict__ row,
                              const long long* __restrict__ col,
                              const float* __restrict__ dinv,
                              const float* __restrict__ src,
                              float* __restrict__ dst, int nEdges) {
    const int wid  = (int)((blockIdx.x * (size_t)blockDim.x + threadIdx.x) >> 5);
    const int lane = threadIdx.x & 31;
    if (wid >= nEdges) return;

    const int r = (int)row[wid];
    const int c = (int)col[wid];
    const float nrm = dinv[r] * dinv[c];

    if constexpr (DIM == 128) {
        float4 v = ((const float4*)(src + (size_t)r * DIM))[lane];
        float* d = dst + (size_t)c * DIM + lane * 4;
        unsafeAtomicAdd(d + 0, v.x * nrm);
        unsafeAtomicAdd(d + 1, v.y * nrm);
        unsafeAtomicAdd(d + 2, v.z * nrm);
        unsafeAtomicAdd(d + 3, v.w * nrm);
    } else {
        float2 v = ((const float2*)(src + (size_t)r * DIM))[lane];
        float* d = dst + (size_t)c * DIM + lane * 2;
        unsafeAtomicAdd(d + 0, v.x * nrm);
        unsafeAtomicAdd(d + 1, v.y * nrm);
    }
}

// ------------------------- self loop + bias (+ relu) -----------------------
// Runs after scatter (stream-ordered), so plain read-modify-write is safe.
template <int DIM, bool RELU>
__global__ void finalize(const float* __restrict__ selfsrc,
                         const float* __restrict__ bias,
                         const float* __restrict__ dinv,
                         float* __restrict__ dst, int nNodes) {
    int idx = blockIdx.x * blockDim.x + threadIdx.x;
    if (idx >= nNodes * DIM) return;
    const int n = idx / DIM;
    const int d = idx % DIM;
    const float di = dinv[n];
    float v = dst[idx] + selfsrc[idx] * (di * di) + bias[d];
    if (RELU) v = fmaxf(v, 0.0f);
    dst[idx] = v;
}

// ---------------------------------- launcher -------------------------------

extern "C" void kernel_launch(void* const* d_in, const int* in_sizes, int n_in,
                              void* d_out, int out_size, void* d_ws, size_t ws_size,
                              hipStream_t stream) {
    const float*     x   = (const float*)d_in[0];
    const long long* ei  = (const long long*)d_in[1];   // int64 [2, E]
    const float*     W1  = (const float*)d_in[2];
    const float*     b1  = (const float*)d_in[3];
    const float*     W2  = (const float*)d_in[4];
    const float*     b2  = (const float*)d_in[5];
    float*           out = (float*)d_out;

    const int E = in_sizes[1] / 2;
    const long long* rowp = ei;
    const long long* colp = ei + E;

    // workspace layout (~103 MB): dinv | bufA (N*128) | bufB (N*128)
    char*  ws   = (char*)d_ws;
    float* dinv = (float*)ws;
    float* bufA = (float*)(ws + (1 << 19));
    float* bufB = (float*)(ws + (1 << 19) + (size_t)N_NODES * HID_DIM * sizeof(float));

    const int T = 256;
    const int nodeBlocks   = (N_NODES + T - 1) / T;
    const int edgeBlocks   = (E + T - 1) / T;
    const int edgeWaveBlks = (int)(((size_t)E * 32 + T - 1) / T);
    const int hidElems     = N_NODES * HID_DIM;
    const int embElems     = N_NODES * EMB_DIM;

    // ---- normalization ----
    init_deg<<<nodeBlocks, T, 0, stream>>>(dinv, N_NODES);
    edge_deg<<<edgeBlocks, T, 0, stream>>>(colp, dinv, E);
    deg_to_dinv<<<nodeBlocks, T, 0, stream>>>(dinv, N_NODES);

    // ---- layer 1: x@W1 -> bufA ; scatter -> bufB ; relu(bufB + self + b1) ----
    gemm_wmma_f32<IN_DIM, HID_DIM><<<N_NODES / 16, 32 * (HID_DIM / 16), 0, stream>>>(x, W1, bufA);
    zero_f32<<<(hidElems + T - 1) / T, T, 0, stream>>>(bufB, hidElems);
    scatter_edges<HID_DIM><<<edgeWaveBlks, T, 0, stream>>>(rowp, colp, dinv, bufA, bufB, E);
    finalize<HID_DIM, true><<<(hidElems + T - 1) / T, T, 0, stream>>>(bufA, b1, dinv, bufB, N_NODES);

    // ---- layer 2: bufB@W2 -> bufA ; scatter -> out ; out + self + b2 ----
    gemm_wmma_f32<HID_DIM, EMB_DIM><<<N_NODES / 16, 32 * (EMB_DIM / 16), 0, stream>>>(bufB, W2, bufA);
    zero_f32<<<(embElems + T - 1) / T, T, 0, stream>>>(out, embElems);
    scatter_edges<EMB_DIM><<<edgeWaveBlks, T, 0, stream>>>(rowp, colp, dinv, bufA, out, E);
    finalize<EMB_DIM, false><<<(embElems + T - 1) / T, T, 0, stream>>>(bufA, b2, dinv, out, N_NODES);
}